// ChamferLoss_Single_66022237274633
// MI455X (gfx1250) — compile-verified
//
#include <hip/hip_runtime.h>
#include <hip/hip_bf16.h>

typedef __attribute__((ext_vector_type(2))) float v2f;
typedef __attribute__((ext_vector_type(8))) float v8f;

#define BATCH 4
#define MPTS  8192
#define NPTS  8192
#define WAVES 8      // waves per block (256 threads, wave32)
#define CHUNK 1024   // dst points staged in LDS per pass (16 KB of float4)

// Zero the scalar output before accumulation (harness poisons d_out).
__global__ void chamfer_init_kernel(float* out) {
    if (threadIdx.x == 0 && blockIdx.x == 0) out[0] = 0.0f;
}

// One wave per 16-row m-tile of one batch. For each 16-col n-tile:
//   D = A(16x4) x B(4x16) via V_WMMA_F32_16X16X4_F32
//   A row m = (x, y, z, 1);  B col n = (-2x, -2y, -2z, ||b||^2)
//   => D[m][n] = -2 a.b + ||b||^2 ; add ||a||^2 after the min over n.
__global__ __launch_bounds__(256)
void ChamferLoss_Single_66022237274633_kernel(const float* __restrict__ src,
                                              const float* __restrict__ dst,
                                              float* __restrict__ out) {
    __shared__ float4 spack[CHUNK];

    const int tid  = threadIdx.x;
    const int wave = tid >> 5;
    const int lane = tid & 31;
    const int half = lane >> 4;   // 0: K=0,1 ; 1: K=2,3
    const int lm   = lane & 15;   // M (for A) / N (for B) within tile

    const int blocksPerBatch = (MPTS / 16) / WAVES;          // 64
    const int b    = blockIdx.x / blocksPerBatch;
    const int mblk = blockIdx.x % blocksPerBatch;
    const int m0   = (mblk * WAVES + wave) * 16;             // tile row base

    const float* srcb = src + (size_t)b * 3 * MPTS;
    const float* dstb = dst + (size_t)b * 3 * NPTS;

    // Loop-invariant A fragment (16x4 f32): lane -> M=lm, VGPRs -> K=2*half+{0,1}
    const int m = m0 + lm;
    v2f afrag;
    if (half == 0) { afrag.x = srcb[0 * MPTS + m]; afrag.y = srcb[1 * MPTS + m]; }
    else           { afrag.x = srcb[2 * MPTS + m]; afrag.y = 1.0f; }

    v8f dmin;
#pragma unroll
    for (int i = 0; i < 8; ++i) dmin[i] = 3.402823466e38f;

    for (int c0 = 0; c0 < NPTS; c0 += CHUNK) {
        // Cooperatively pack dst points: (-2x, -2y, -2z, ||b||^2)
        for (int j = tid; j < CHUNK; j += 256) {
            const int n = c0 + j;
            const float x = dstb[0 * NPTS + n];
            const float y = dstb[1 * NPTS + n];
            const float z = dstb[2 * NPTS + n];
            spack[j] = make_float4(-2.0f * x, -2.0f * y, -2.0f * z,
                                   x * x + y * y + z * z);
        }
        __syncthreads();

#pragma unroll 4
        for (int nt = 0; nt < CHUNK; nt += 16) {
            // B fragment (4x16 f32): lane -> N=lm, VGPRs -> K=2*half+{0,1}
            const v2f bfrag =
                *(const v2f*)((const float*)&spack[nt + lm] + 2 * half);
            v8f czero = {};
            v8f d = __builtin_amdgcn_wmma_f32_16x16x4_f32(
                /*neg_a=*/false, afrag, /*neg_b=*/false, bfrag,
                /*c_mod=*/(short)0, czero, /*reuse_a=*/false, /*reuse_b=*/false);
#pragma unroll
            for (int i = 0; i < 8; ++i) dmin[i] = fminf(dmin[i], d[i]);
        }
        __syncthreads();
    }

    // Min across the 16 lanes of each half-wave (min over n within a row).
#pragma unroll
    for (int off = 1; off < 16; off <<= 1) {
#pragma unroll
        for (int i = 0; i < 8; ++i) {
            const float o = __shfl_xor(dmin[i], off, 32);
            dmin[i] = fminf(dmin[i], o);
        }
    }

    // D layout: VGPR i of half h holds row m0 + 8*h + i. Lanes lm==0 finalize.
    if (lm == 0) {
        float s = 0.0f;
#pragma unroll
        for (int i = 0; i < 8; ++i) {
            const int   mr = m0 + half * 8 + i;
            const float x  = srcb[0 * MPTS + mr];
            const float y  = srcb[1 * MPTS + mr];
            const float z  = srcb[2 * MPTS + mr];
            const float d2 = dmin[i] + (x * x + y * y + z * z);
            s += sqrtf(fmaxf(d2, 0.0f));
        }
        atomicAdd(out, s * (1.0f / ((float)BATCH * (float)MPTS)));
    }
}

extern "C" void kernel_launch(void* const* d_in, const int* in_sizes, int n_in,
                              void* d_out, int out_size, void* d_ws, size_t ws_size,
                              hipStream_t stream) {
    const float* src = (const float*)d_in[0];  // [B, 3, M]
    const float* dst = (const float*)d_in[1];  // [B, 3, N]
    float* out = (float*)d_out;                // scalar mean

    hipLaunchKernelGGL(chamfer_init_kernel, dim3(1), dim3(1), 0, stream, out);

    const int grid = BATCH * (MPTS / 16) / WAVES;  // 256 blocks
    hipLaunchKernelGGL(ChamferLoss_Single_66022237274633_kernel,
                       dim3(grid), dim3(256), 0, stream, src, dst, out);
}